// SliceLSTMConv_62113817035361
// MI455X (gfx1250) — compile-verified
//
#include <hip/hip_runtime.h>

// ---------------------------------------------------------------------------
// SliceLSTMConv for MI455X (gfx1250, wave32)
//
// Stage 1: transpose x -> xT (coalesced time-rows for the x-axis scan)
// Stage 2: 64 workgroups = {axis(2) x dir(2) x batch(16)}, 512 threads each.
//          Per step: load 3 input channels, publish z=[c0,c1,c2,h] to a
//          double-buffered LDS plane, ONE barrier, then compute the 4 gate
//          pre-activations as an fp32 WMMA GEMM
//              G^T(16x16) = sum_tap A_tap(16x4 = Wg, gates padded) x
//                            B_tap(4x16 = z at positions p+tap-1)  + bias(C)
//          using V_WMMA_F32_16X16X4_F32, then the pointwise LSTM update
//          (activations via v_exp_f32 + v_rcp_f32, no IEEE divides).
// Stage 3: 3x3 conv over [res_x, res_y] + sigmoid -> out.
// ---------------------------------------------------------------------------

#define SDIM 512
#define SPAD 514   // z plane padded with a zero column on each end

typedef __attribute__((ext_vector_type(2))) float v2f;
typedef __attribute__((ext_vector_type(8))) float v8f;

#if defined(__gfx1250__) && __has_builtin(__builtin_amdgcn_wmma_f32_16x16x4_f32)
#define USE_WMMA 1
#else
#define USE_WMMA 0
#endif

__device__ __forceinline__ float fast_rcp(float v) {
#if defined(__AMDGCN__) && __has_builtin(__builtin_amdgcn_rcpf)
    return __builtin_amdgcn_rcpf(v);   // v_rcp_f32, ~1 ulp
#else
    return 1.0f / v;
#endif
}
__device__ __forceinline__ float fast_sig(float v) {
    return fast_rcp(1.0f + __expf(-v));
}
__device__ __forceinline__ float fast_tanh(float v) {
    // tanh(x) = 1 - 2/(exp(2x)+1); exact at +/-inf, fp32-accurate in between
    return 1.0f - 2.0f * fast_rcp(__expf(2.0f * v) + 1.0f);
}

// ---------------------------------------------------------------- transpose
__global__ __launch_bounds__(1024) void transpose_kernel(
    const float* __restrict__ x, float* __restrict__ xT) {
    __shared__ float tile[32][33];
    const int b  = blockIdx.z;
    const int h0 = blockIdx.y * 32;
    const int w0 = blockIdx.x * 32;
    const float* xb  = x  + (size_t)b * SDIM * SDIM;
    float*       xtb = xT + (size_t)b * SDIM * SDIM;
    const int tx = threadIdx.x, ty = threadIdx.y;
    tile[ty][tx] = xb[(size_t)(h0 + ty) * SDIM + (w0 + tx)];
    __syncthreads();
    xtb[(size_t)(w0 + ty) * SDIM + (h0 + tx)] = tile[tx][ty];
}

// --------------------------------------------------------------------- scan
__global__ __launch_bounds__(512) void scan_kernel(
    const float* __restrict__ x,      // [16][512][512] row-major (h,w)
    const float* __restrict__ xT,     // [16][512][512] transposed (w,h)
    const float* __restrict__ Wg,     // [4 out][4 in][3 tap]
    const float* __restrict__ bg,     // [4]
    float* __restrict__ res_xT,       // [16][w][h]  (x-scan result, transposed)
    float* __restrict__ res_y) {      // [16][h][w]  (y-scan result)
    __shared__ float zs[2][4][SPAD];  // double-buffered z planes: [buf][ch][pos+1]
    __shared__ float gs[4][SDIM];     // gate staging (per-wave private slices)

    const int s    = threadIdx.x;           // spatial position, 0..511
    const int wg   = blockIdx.x;
    const int b    = wg & 15;
    const int dir  = (wg >> 4) & 1;         // 0: middle->0 (backward), 1: middle->511
    const int axis = (wg >> 5) & 1;         // 0: x-axis scan, 1: y-axis scan

    const float* base  = ((axis == 0) ? xT : x) + (size_t)b * SDIM * SDIM;
    float*       obase = ((axis == 0) ? res_xT : res_y) + (size_t)b * SDIM * SDIM;

    const int lane = s & 31;
    const int hi   = lane >> 4;     // which half of the wave
    const int nn   = lane & 15;     // tile-local position / matrix row index
    const int wave = s >> 5;

#if USE_WMMA
    // A_tap: 16x4 fp32 matrix = Wg for one kernel tap; rows(M)=gates (4 used,
    // 12 zero-padded), cols(K)=input channels. ISA layout: lanes 0-15 hold
    // K=0,1 in VGPR0,1; lanes 16-31 hold K=2,3.
    v2f Apre[3];
    #pragma unroll
    for (int tap = 0; tap < 3; ++tap) {
        Apre[tap].x = (nn < 4) ? Wg[nn * 12 + (2 * hi + 0) * 3 + tap] : 0.0f;
        Apre[tap].y = (nn < 4) ? Wg[nn * 12 + (2 * hi + 1) * 3 + tap] : 0.0f;
    }
    // Bias preloaded into C: D[m=gate][n=pos] layout -> lanes 0-15 carry rows
    // M=0..7 in VGPRs 0..7 (only M<4 used), lanes 16-31 carry rows 8..15 (zero).
    v8f Cb;
    #pragma unroll
    for (int r = 0; r < 8; ++r) Cb[r] = (hi == 0 && r < 4) ? bg[r] : 0.0f;
#else
    float Wr[48];
    #pragma unroll
    for (int i = 0; i < 48; ++i) Wr[i] = Wg[i];
    float bgr[4];
    #pragma unroll
    for (int i = 0; i < 4; ++i) bgr[i] = bg[i];
#endif

    // zero the pad columns of both z buffers once
    if (s < 16) {
        const int bf = s >> 3, ch = (s >> 1) & 3, e = s & 1;
        zs[bf][ch][e ? (SPAD - 1) : 0] = 0.0f;
    }
    __syncthreads();

    float h = 0.0f, c = 0.0f;
    const int nsteps = dir ? 256 : 257;
    for (int k = 0; k < nsteps; ++k) {
        const int t  = dir ? (256 + k) : (256 - k);  // time coordinate
        const int bf = k & 1;

        // ---- gather the 3 input channels (reflect pad along w) ----
        float c0, c1, c2;
        if (axis == 0) {
            const int tm = (t == 0) ? 1 : t - 1;
            const int tp = (t == SDIM - 1) ? SDIM - 2 : t + 1;
            c1 = base[(size_t)t  * SDIM + s];
            c0 = base[(size_t)tm * SDIM + s];
            c2 = base[(size_t)tp * SDIM + s];
            const int tn = dir ? min(t + 2, SDIM - 1) : max(t - 2, 0);
            __builtin_prefetch(&base[(size_t)tn * SDIM + s], 0, 1);
        } else {
            const int sm = (s == 0) ? 1 : s - 1;
            const int sp = (s == SDIM - 1) ? SDIM - 2 : s + 1;
            const float* row = base + (size_t)t * SDIM;
            c1 = row[s];
            c0 = row[sm];
            c2 = row[sp];
        }

        // ---- publish z = [c0, c1, c2, h_prev] ----
        zs[bf][0][s + 1] = c0;
        zs[bf][1][s + 1] = c1;
        zs[bf][2][s + 1] = c2;
        zs[bf][3][s + 1] = h;
        __syncthreads();   // single barrier per step (double-buffered z)

        // ---- gate pre-activations: G = Wg (*) z, kernel 3, zero pad ----
        float g0, g1, g2, g3;
#if USE_WMMA
        {
            #pragma unroll
            for (int half = 0; half < 2; ++half) {
                const int tb = wave * 32 + half * 16;   // 16-position tile base
                v8f acc = Cb;
                #pragma unroll
                for (int tap = 0; tap < 3; ++tap) {
                    // B_tap: 4x16 fp32, rows(K)=channels, cols(N)=positions,
                    // tap shift folded into the (zero-padded) LDS index.
                    v2f Bv;
                    Bv.x = zs[bf][2 * hi + 0][tb + nn + tap];
                    Bv.y = zs[bf][2 * hi + 1][tb + nn + tap];
                    acc = __builtin_amdgcn_wmma_f32_16x16x4_f32(
                        false, Apre[tap], false, Bv, (short)0, acc,
                        false, false);
                }
                // D: lane n (0..15) holds gates 0..3 for position tb+n in
                // VGPRs 0..3 -> short LDS hop back to one-position-per-lane.
                if (hi == 0) {
                    gs[0][tb + nn] = acc[0];
                    gs[1][tb + nn] = acc[1];
                    gs[2][tb + nn] = acc[2];
                    gs[3][tb + nn] = acc[3];
                }
            }
            // same-wave producer/consumer: no workgroup barrier needed
            g0 = gs[0][s]; g1 = gs[1][s]; g2 = gs[2][s]; g3 = gs[3][s];
        }
#else
        {
            float acc[4] = {bgr[0], bgr[1], bgr[2], bgr[3]};
            #pragma unroll
            for (int i = 0; i < 4; ++i) {
                const float zm = zs[bf][i][s];
                const float zc = zs[bf][i][s + 1];
                const float zp = zs[bf][i][s + 2];
                #pragma unroll
                for (int o = 0; o < 4; ++o)
                    acc[o] += Wr[o * 12 + i * 3 + 0] * zm +
                              Wr[o * 12 + i * 3 + 1] * zc +
                              Wr[o * 12 + i * 3 + 2] * zp;
            }
            g0 = acc[0]; g1 = acc[1]; g2 = acc[2]; g3 = acc[3];
        }
#endif

        // ---- pointwise LSTM update (exp + rcp, no IEEE division) ----
        const float ig = fast_sig(g0);
        const float fg = fast_sig(g1);
        const float og = fast_sig(g2);
        const float gg = fast_tanh(g3);
        c = fg * c + ig * gg;
        h = og * fast_tanh(c);

        // backward scan drops its first (t==256) output
        if (dir == 1 || k > 0) obase[(size_t)t * SDIM + s] = h;
    }
}

// --------------------------------------------------------------- final conv
__global__ __launch_bounds__(256) void outconv_kernel(
    const float* __restrict__ res_xT,  // [16][w][h]
    const float* __restrict__ res_y,   // [16][h][w]
    const float* __restrict__ Wc,      // [1][2][3][3]
    const float* __restrict__ bc,      // [1]
    float* __restrict__ outp) {        // [16][h][w]
    const int idx = blockIdx.x * 256 + threadIdx.x;
    const int w = idx & (SDIM - 1);
    const int h = (idx >> 9) & (SDIM - 1);
    const int b = idx >> 18;
    const float* rx = res_xT + (size_t)b * SDIM * SDIM;
    const float* ry = res_y  + (size_t)b * SDIM * SDIM;
    float acc = bc[0];
    #pragma unroll
    for (int ky = 0; ky < 3; ++ky) {
        const int hh = h + ky - 1;
        if (hh < 0 || hh >= SDIM) continue;
        #pragma unroll
        for (int kx = 0; kx < 3; ++kx) {
            const int ww = w + kx - 1;
            if (ww < 0 || ww >= SDIM) continue;
            const float vx = rx[(size_t)ww * SDIM + hh];  // res_x[b][hh][ww]
            const float vy = ry[(size_t)hh * SDIM + ww];
            acc += Wc[ky * 3 + kx] * vx + Wc[9 + ky * 3 + kx] * vy;
        }
    }
    outp[idx] = fast_sig(acc);
}

// ------------------------------------------------------------------- launch
extern "C" void kernel_launch(void* const* d_in, const int* in_sizes, int n_in,
                              void* d_out, int out_size, void* d_ws, size_t ws_size,
                              hipStream_t stream) {
    const float* x  = (const float*)d_in[0];   // [16][512][512]
    const float* Wg = (const float*)d_in[1];   // [4][4][3]
    const float* bg = (const float*)d_in[2];   // [4]
    const float* Wc = (const float*)d_in[3];   // [1][2][3][3]
    const float* bc = (const float*)d_in[4];   // [1]
    float* outp = (float*)d_out;

    const size_t N = (size_t)16 * SDIM * SDIM;
    float* xT     = (float*)d_ws;
    float* res_xT = xT + N;
    float* res_y  = xT + 2 * N;

    transpose_kernel<<<dim3(SDIM / 32, SDIM / 32, 16), dim3(32, 32), 0, stream>>>(x, xT);
    scan_kernel<<<64, 512, 0, stream>>>(x, xT, Wg, bg, res_xT, res_y);
    outconv_kernel<<<(int)(N / 256), 256, 0, stream>>>(res_xT, res_y, Wc, bc, outp);
}